// MultiScaleDeformableAttention_68607807586916
// MI455X (gfx1250) — compile-verified
//
#include <hip/hip_runtime.h>
#include <hip/hip_bf16.h>
#include <math.h>

typedef __attribute__((ext_vector_type(2))) float v2f;
typedef __attribute__((ext_vector_type(8))) float v8f;

#define WPB 8      // waves per block (block = 32 x 8)
#define KMAX 256   // K is 256 for every GEMM in this op
#define BROW 40    // LDS row stride (dwords): 2*40 = 80 = 16 mod 64 -> the two
                   // half-waves (rows k and k+2) hit disjoint 16-bank ranges

// ---------------------------------------------------------------------------
// WMMA f32 GEMM:  C[M,N] = A[M,K] @ B[K,N] + bias[N] (+ residual[M,N])
// Block: 8 waves, all sharing one 32-column B panel staged in LDS.
// Each wave computes a 16x32 tile = two V_WMMA_F32_16X16X4_F32 accumulators,
// reusing a single A fetch per k-step (1 global b64 + 4 ds b32 + 2 WMMA).
// A layout per ISA 7.12.2 (32-bit A 16x4): lanes 0-15 = rows, half = K pair.
// B mirrored: lanes 0-15 = cols, half = K pair. C/D: 8 VGPRs, row = i + half*8.
// ---------------------------------------------------------------------------
__global__ __launch_bounds__(256) void msda_wmma_gemm(
    const float* __restrict__ A, const float* __restrict__ B,
    const float* __restrict__ bias, const float* __restrict__ residual,
    float* __restrict__ C, int M, int N, int K)
{
    __shared__ float Bs[KMAX * BROW];   // 40 KB

    const int lane = threadIdx.x;            // 0..31
    const int wave = threadIdx.y;            // 0..7
    const int tid  = wave * 32 + lane;       // 0..255
    const int col0 = blockIdx.y * 32;

    // Cooperative stage of B[K][col0..col0+31] into LDS (row-major, padded).
    for (int idx = tid; idx < K * 32; idx += 256) {
        const int kk = idx >> 5;
        const int cc = idx & 31;
        Bs[kk * BROW + cc] = B[(size_t)kk * N + col0 + cc];
    }
    __syncthreads();

    const int tileM = blockIdx.x * WPB + wave;
    const int row0 = tileM * 16;
    if (row0 >= M) return;                   // uniform per wave, after barrier

    const int ln   = lane & 15;
    const int half = lane >> 4;              // 0: K=0,1   1: K=2,3

    int arow = row0 + ln;
    if (arow >= M) arow = M - 1;             // clamp loads; stores are guarded
    const float* Ap  = A  + (size_t)arow * K + (half << 1);
    const float* Bp0 = Bs + (half << 1) * BROW + ln;        // tile-0 column
    const float* Bp1 = Bp0 + 16;                            // tile-1 column

    const float bv0 = bias[col0 + ln];
    const float bv1 = bias[col0 + 16 + ln];
    v8f c0, c1;
#pragma unroll
    for (int i = 0; i < 8; ++i) { c0[i] = bv0; c1[i] = bv1; }

    for (int k = 0; k < K; k += 4) {
        v2f a, b0, b1;
        a[0]  = Ap[k];
        a[1]  = Ap[k + 1];
        b0[0] = Bp0[k * BROW];
        b0[1] = Bp0[(k + 1) * BROW];
        b1[0] = Bp1[k * BROW];
        b1[1] = Bp1[(k + 1) * BROW];
        c0 = __builtin_amdgcn_wmma_f32_16x16x4_f32(
                false, a, false, b0, (short)0, c0, false, false);
        c1 = __builtin_amdgcn_wmma_f32_16x16x4_f32(
                false, a, false, b1, (short)0, c1, false, false);
    }

#pragma unroll
    for (int i = 0; i < 8; ++i) {
        const int r = row0 + (half << 3) + i;
        if (r < M) {
            const size_t idx0 = (size_t)r * N + col0 + ln;
            float v0 = c0[i], v1 = c1[i];
            if (residual) { v0 += residual[idx0]; v1 += residual[idx0 + 16]; }
            C[idx0]      = v0;
            C[idx0 + 16] = v1;
        }
    }
}

// ---------------------------------------------------------------------------
// Softmax over 16 contiguous logits per (b,q,h) row, in place.
// ---------------------------------------------------------------------------
__global__ __launch_bounds__(256) void msda_softmax16(float* __restrict__ x, int nrows)
{
    const int r = blockIdx.x * blockDim.x + threadIdx.x;
    if (r >= nrows) return;
    float* p = x + (size_t)r * 16;
    float v[16];
    float m = -3.402823466e38f;
#pragma unroll
    for (int i = 0; i < 16; ++i) { v[i] = p[i]; m = fmaxf(m, v[i]); }
    float s = 0.f;
#pragma unroll
    for (int i = 0; i < 16; ++i) { v[i] = __expf(v[i] - m); s += v[i]; }
    const float inv = 1.0f / s;
#pragma unroll
    for (int i = 0; i < 16; ++i) p[i] = v[i] * inv;
}

// ---------------------------------------------------------------------------
// Bilinear sampling + attention-weighted accumulation.
// One wave per (b,q,h); lane = channel d in [0,32). All addressing is wave-
// uniform; each corner gather is a contiguous 128B wave load from v.
// ---------------------------------------------------------------------------
__global__ __launch_bounds__(256) void msda_sample(
    const float* __restrict__ v,       // (bs*nv, 256)  channels h*32+d
    const float* __restrict__ so,      // (bs*nq, 256)  = (nh,nl,np,2)
    const float* __restrict__ aw,      // (bs*nq, 128)  = (nh,nl,np)
    const float* __restrict__ refp,    // (bs*nq, NL, 2)
    const int* __restrict__ shapes,    // (NL, 2) = (h, w)
    const int* __restrict__ starts,    // (NL,)
    float* __restrict__ out,           // (bs*nq, 256)
    int nq, int nv, int total_waves)
{
    const int wid = blockIdx.x * WPB + threadIdx.y;
    if (wid >= total_waves) return;
    const int lane = threadIdx.x;      // channel within head
    const int h  = wid & 7;
    const int bq = wid >> 3;           // b*nq + q
    const int b  = bq / nq;

    const float* so_row  = so  + (size_t)bq * 256 + h * 32;  // 16 (x,y) pairs
    const float* aw_row  = aw  + (size_t)bq * 128 + h * 16;  // 16 weights
    const float* ref_row = refp + (size_t)bq * 8;            // NL * 2

    float acc = 0.0f;
#pragma unroll
    for (int l = 0; l < 4; ++l) {
        const int H = shapes[l * 2 + 0];
        const int W = shapes[l * 2 + 1];
        const int s = starts[l];
        const float rx = ref_row[l * 2 + 0];
        const float ry = ref_row[l * 2 + 1];
        const float* vb = v + ((size_t)b * nv + s) * 256 + h * 32 + lane;
        const float fW = (float)W, fH = (float)H;
#pragma unroll
        for (int p = 0; p < 4; ++p) {
            const float sox = so_row[(l * 4 + p) * 2 + 0];
            const float soy = so_row[(l * 4 + p) * 2 + 1];
            const float x = (rx + sox / fW) * fW - 0.5f;
            const float y = (ry + soy / fH) * fH - 0.5f;
            const float fx0 = floorf(x), fy0 = floorf(y);
            const int x0 = (int)fx0, y0 = (int)fy0;
            const int x1 = x0 + 1,   y1 = y0 + 1;
            const float wx1 = x - fx0, wx0 = 1.0f - wx1;
            const float wy1 = y - fy0, wy0 = 1.0f - wy1;
            const bool vx0 = (x0 >= 0) & (x0 < W);
            const bool vx1 = (x1 >= 0) & (x1 < W);
            const bool vy0 = (y0 >= 0) & (y0 < H);
            const bool vy1 = (y1 >= 0) & (y1 < H);
            float g = 0.0f;
            if (vx0 & vy0) g += wx0 * wy0 * vb[(size_t)(y0 * W + x0) * 256];
            if (vx1 & vy0) g += wx1 * wy0 * vb[(size_t)(y0 * W + x1) * 256];
            if (vx0 & vy1) g += wx0 * wy1 * vb[(size_t)(y1 * W + x0) * 256];
            if (vx1 & vy1) g += wx1 * wy1 * vb[(size_t)(y1 * W + x1) * 256];
            acc += aw_row[l * 4 + p] * g;
        }
    }
    out[(size_t)bq * 256 + h * 32 + lane] = acc;
}

extern "C" void kernel_launch(void* const* d_in, const int* in_sizes, int n_in,
                              void* d_out, int out_size, void* d_ws, size_t ws_size,
                              hipStream_t stream) {
    (void)n_in; (void)out_size; (void)ws_size;
    const float* query = (const float*)d_in[0];
    const float* value = (const float*)d_in[1];
    const float* refp  = (const float*)d_in[2];
    const int*   shp   = (const int*)d_in[3];
    const int*   strt  = (const int*)d_in[4];
    const float* Wv    = (const float*)d_in[5];
    const float* bv    = (const float*)d_in[6];
    const float* Wso   = (const float*)d_in[7];
    const float* bso   = (const float*)d_in[8];
    const float* Waw   = (const float*)d_in[9];
    const float* baw   = (const float*)d_in[10];
    const float* Wo    = (const float*)d_in[11];
    const float* bo    = (const float*)d_in[12];
    float* out = (float*)d_out;

    const int BS_ = 2, E_ = 256, NH_ = 8;
    const int nq = in_sizes[0] / (BS_ * E_);
    const int nv = in_sizes[1] / (BS_ * E_);
    const int M  = BS_ * nq;   // query-side rows
    const int Mv = BS_ * nv;   // value-side rows

    float* ws      = (float*)d_ws;
    float* ws_v    = ws;                              // Mv * 256
    float* ws_so   = ws_v  + (size_t)Mv * E_;         // M  * 256
    float* ws_aw   = ws_so + (size_t)M  * E_;         // M  * 128 (softmax in place)
    float* ws_attn = ws_aw + (size_t)M  * 128;        // M  * 256

    const dim3 blk(32, WPB, 1);

    // v = value @ Wv + bv
    {
        dim3 grd((Mv + 16 * WPB - 1) / (16 * WPB), E_ / 32, 1);
        msda_wmma_gemm<<<grd, blk, 0, stream>>>(value, Wv, bv, nullptr, ws_v, Mv, E_, E_);
    }
    // so = query @ Wso + bso
    {
        dim3 grd((M + 16 * WPB - 1) / (16 * WPB), 256 / 32, 1);
        msda_wmma_gemm<<<grd, blk, 0, stream>>>(query, Wso, bso, nullptr, ws_so, M, 256, E_);
    }
    // logits = query @ Waw + baw
    {
        dim3 grd((M + 16 * WPB - 1) / (16 * WPB), 128 / 32, 1);
        msda_wmma_gemm<<<grd, blk, 0, stream>>>(query, Waw, baw, nullptr, ws_aw, M, 128, E_);
    }
    // softmax over the 16 (l,p) logits per (b,q,h)
    {
        const int rows = M * NH_;
        msda_softmax16<<<(rows + 255) / 256, 256, 0, stream>>>(ws_aw, rows);
    }
    // bilinear sampling + weighted accumulation
    {
        const int waves = M * NH_;
        msda_sample<<<(waves + WPB - 1) / WPB, blk, 0, stream>>>(
            ws_v, ws_so, ws_aw, refp, shp, strt, ws_attn, nq, nv, waves);
    }
    // out = attn @ Wo + bo + query   (residual fused)
    {
        dim3 grd((M + 16 * WPB - 1) / (16 * WPB), 256 / 32, 1);
        msda_wmma_gemm<<<grd, blk, 0, stream>>>(ws_attn, Wo, bo, query, out, M, 256, E_);
    }
}